// Mamba_8177617731785
// MI455X (gfx1250) — compile-verified
//
#include <hip/hip_runtime.h>
#include <hip/hip_bf16.h>

typedef __attribute__((ext_vector_type(2))) float v2f;
typedef __attribute__((ext_vector_type(8))) float v8f;

#define D_MODEL 768
#define D_STATE 16
#define D_CONV  4
#define D_INNER 1536
#define DT_RANK 48
#define B_SZ    2
#define SEQ     1024
#define BL      (B_SZ * SEQ)            // 2048 rows
#define XPROJ_N (DT_RANK + 2 * D_STATE) // 80

#define WMMA_F32(acc, a, b)                                                  \
    acc = __builtin_amdgcn_wmma_f32_16x16x4_f32(false, (a), false, (b),      \
                                                (short)0, (acc), false, false)

// ---------------------------------------------------------------------------
// C(M,N) = A(M,K; lda) * W(N,K; ldw)^T  via V_WMMA_F32_16X16X4_F32 (wave32)
// Block = 128 threads (4 waves). Each wave computes a 32(M) x 64(N) strip:
// 8 statically-named 16x16 f32 accumulators (2 M-tiles x 4 N-tiles), so each
// k-step is 8 independent WMMAs fed by 6 float2 loads (2 A rows + 4 W rows).
// Distance-2 software pipeline: fragments for k-steps {k0, k0+4} are resident
// while the pair {k0+8, k0+12} is prefetched, so s_wait_loadcnt covers an
// entire 16-WMMA group of latency. K is a multiple of 8 for every call here
// (768 / 1536 / 48), so no remainder path is needed.
//
// f32 16x16x4 fragment layout (ISA 7.12.2):
//   A: lane holds A[m = lane&15][k0 + 2*(lane>>4) + g], g in {0,1}  (float2)
//   B: lane holds B[k0 + 2*(lane>>4) + g][n = lane&15] = W[n][k...]  (float2)
//   C/D: vgpr g holds C[m0 + g + 8*(lane>>4)][n0 + (lane&15)]
// ---------------------------------------------------------------------------
__global__ __launch_bounds__(128) void wmma_gemm_nt(
    const float* __restrict__ A, int lda,
    const float* __restrict__ W, int ldw,
    float* __restrict__ C, int ldc,
    int N, int K)
{
    const int lane = threadIdx.x & 31;
    const int wave = threadIdx.x >> 5;      // 0..3
    const int half = lane >> 4;             // 0 or 1
    const int l16  = lane & 15;
    const int m0   = blockIdx.y << 5;       // 32 rows per block-row
    const int n0   = blockIdx.x * 256 + wave * 64;
    if (n0 >= N) return;                    // wave-uniform

    const float* arow0 = A + (size_t)(m0 + l16) * lda + 2 * half;
    const float* arow1 = arow0 + (size_t)16 * lda;

    int nc0 = n0 + 0  + l16; if (nc0 > N - 1) nc0 = N - 1;  // clamp: no OOB
    int nc1 = n0 + 16 + l16; if (nc1 > N - 1) nc1 = N - 1;
    int nc2 = n0 + 32 + l16; if (nc2 > N - 1) nc2 = N - 1;
    int nc3 = n0 + 48 + l16; if (nc3 > N - 1) nc3 = N - 1;
    const float* wrow0 = W + (size_t)nc0 * ldw + 2 * half;
    const float* wrow1 = W + (size_t)nc1 * ldw + 2 * half;
    const float* wrow2 = W + (size_t)nc2 * ldw + 2 * half;
    const float* wrow3 = W + (size_t)nc3 * ldw + 2 * half;

    v8f acc00 = {}, acc01 = {}, acc02 = {}, acc03 = {};   // M-tile 0
    v8f acc10 = {}, acc11 = {}, acc12 = {}, acc13 = {};   // M-tile 1

    // Prologue: two pipeline stages (k = 0 and k = 4)
    v2f a0A = *(const v2f*)(arow0);     v2f a0B = *(const v2f*)(arow0 + 4);
    v2f a1A = *(const v2f*)(arow1);     v2f a1B = *(const v2f*)(arow1 + 4);
    v2f b0A = *(const v2f*)(wrow0);     v2f b0B = *(const v2f*)(wrow0 + 4);
    v2f b1A = *(const v2f*)(wrow1);     v2f b1B = *(const v2f*)(wrow1 + 4);
    v2f b2A = *(const v2f*)(wrow2);     v2f b2B = *(const v2f*)(wrow2 + 4);
    v2f b3A = *(const v2f*)(wrow3);     v2f b3B = *(const v2f*)(wrow3 + 4);

    for (int k0 = 8; k0 < K; k0 += 8) {
        // prefetch next pair of k-steps before consuming current fragments
        const v2f na0A = *(const v2f*)(arow0 + k0);
        const v2f na1A = *(const v2f*)(arow1 + k0);
        const v2f nb0A = *(const v2f*)(wrow0 + k0);
        const v2f nb1A = *(const v2f*)(wrow1 + k0);
        const v2f nb2A = *(const v2f*)(wrow2 + k0);
        const v2f nb3A = *(const v2f*)(wrow3 + k0);
        const v2f na0B = *(const v2f*)(arow0 + k0 + 4);
        const v2f na1B = *(const v2f*)(arow1 + k0 + 4);
        const v2f nb0B = *(const v2f*)(wrow0 + k0 + 4);
        const v2f nb1B = *(const v2f*)(wrow1 + k0 + 4);
        const v2f nb2B = *(const v2f*)(wrow2 + k0 + 4);
        const v2f nb3B = *(const v2f*)(wrow3 + k0 + 4);

        WMMA_F32(acc00, a0A, b0A);  WMMA_F32(acc01, a0A, b1A);
        WMMA_F32(acc02, a0A, b2A);  WMMA_F32(acc03, a0A, b3A);
        WMMA_F32(acc10, a1A, b0A);  WMMA_F32(acc11, a1A, b1A);
        WMMA_F32(acc12, a1A, b2A);  WMMA_F32(acc13, a1A, b3A);

        WMMA_F32(acc00, a0B, b0B);  WMMA_F32(acc01, a0B, b1B);
        WMMA_F32(acc02, a0B, b2B);  WMMA_F32(acc03, a0B, b3B);
        WMMA_F32(acc10, a1B, b0B);  WMMA_F32(acc11, a1B, b1B);
        WMMA_F32(acc12, a1B, b2B);  WMMA_F32(acc13, a1B, b3B);

        a0A = na0A; a1A = na1A; b0A = nb0A; b1A = nb1A; b2A = nb2A; b3A = nb3A;
        a0B = na0B; a1B = na1B; b0B = nb0B; b1B = nb1B; b2B = nb2B; b3B = nb3B;
    }
    // Epilogue: last two k-steps
    WMMA_F32(acc00, a0A, b0A);  WMMA_F32(acc01, a0A, b1A);
    WMMA_F32(acc02, a0A, b2A);  WMMA_F32(acc03, a0A, b3A);
    WMMA_F32(acc10, a1A, b0A);  WMMA_F32(acc11, a1A, b1A);
    WMMA_F32(acc12, a1A, b2A);  WMMA_F32(acc13, a1A, b3A);
    WMMA_F32(acc00, a0B, b0B);  WMMA_F32(acc01, a0B, b1B);
    WMMA_F32(acc02, a0B, b2B);  WMMA_F32(acc03, a0B, b3B);
    WMMA_F32(acc10, a1B, b0B);  WMMA_F32(acc11, a1B, b1B);
    WMMA_F32(acc12, a1B, b2B);  WMMA_F32(acc13, a1B, b3B);

    #pragma unroll
    for (int g = 0; g < 8; ++g) {
        float* cr0 = C + (size_t)(m0 + 8 * half + g) * ldc + l16;
        float* cr1 = C + (size_t)(m0 + 16 + 8 * half + g) * ldc + l16;
        cr0[n0] = acc00[g];
        cr1[n0] = acc10[g];
        if (n0 + 16 < N) { cr0[n0 + 16] = acc01[g]; cr1[n0 + 16] = acc11[g]; }
        if (n0 + 32 < N) { cr0[n0 + 32] = acc02[g]; cr1[n0 + 32] = acc12[g]; }
        if (n0 + 48 < N) { cr0[n0 + 48] = acc03[g]; cr1[n0 + 48] = acc13[g]; }
    }
}

// ---------------------------------------------------------------------------
// Depthwise causal conv (k=4) + bias + SiLU.
// x lives in the first half of each xz row: xz[(b*SEQ+l)*2*D_INNER + d].
// ---------------------------------------------------------------------------
__global__ __launch_bounds__(256) void conv_silu_kernel(
    const float* __restrict__ xz,
    const float* __restrict__ conv_w,    // (D_INNER, 4)
    const float* __restrict__ conv_b,    // (D_INNER)
    float* __restrict__ xconv)           // (BL, D_INNER)
{
    const int idx = blockIdx.x * 256 + threadIdx.x;   // grid sized exactly
    const int d   = idx % D_INNER;
    const int bl  = idx / D_INNER;
    const int l   = bl % SEQ;
    const int b   = bl / SEQ;

    float acc = conv_b[d];
    #pragma unroll
    for (int k = 0; k < D_CONV; ++k) {
        const int ls = l + k - (D_CONV - 1);
        if (ls >= 0) {
            acc += xz[(size_t)(b * SEQ + ls) * (2 * D_INNER) + d] *
                   conv_w[d * D_CONV + k];
        }
    }
    xconv[idx] = acc / (1.0f + __expf(-acc));   // SiLU
}

// ---------------------------------------------------------------------------
// Fused selective scan. One 16-lane group per (b, d) channel; lane n owns
// state h[n]. Per step: softplus(delta), h = exp(delta*A)*h + delta*u*B,
// y = reduce16(h*C), then skip (u*D) and z-gate SiLU, store gated y.
// ---------------------------------------------------------------------------
__global__ __launch_bounds__(256) void scan_kernel(
    const float* __restrict__ xz,        // (BL, 2*D_INNER): z in 2nd half
    const float* __restrict__ xconv,     // (BL, D_INNER)
    const float* __restrict__ x_dbl,     // (BL, 80): [dt(48) | B(16) | C(16)]
    const float* __restrict__ draw,      // (BL, D_INNER) delta pre-bias
    const float* __restrict__ dt_proj_b, // (D_INNER)
    const float* __restrict__ A_log,     // (D_INNER, D_STATE)
    const float* __restrict__ Dvec,      // (D_INNER)
    float* __restrict__ ybuf)            // (BL, D_INNER)
{
    const int gid   = blockIdx.x * 256 + threadIdx.x;
    const int group = gid >> 4;              // 0 .. B_SZ*D_INNER-1
    const int n     = gid & 15;
    const int b     = group / D_INNER;
    const int d     = group - b * D_INNER;

    const float Adn = -__expf(A_log[d * D_STATE + n]);
    const float dtb = dt_proj_b[d];
    const float Dd  = Dvec[d];

    float h = 0.0f;
    for (int l = 0; l < SEQ; ++l) {
        const int row = b * SEQ + l;
        const float dr = draw[(size_t)row * D_INNER + d];       // broadcast
        const float xb = dr + dtb;
        const float delta = (xb > 20.0f) ? xb : log1pf(__expf(xb));
        const float u  = xconv[(size_t)row * D_INNER + d];      // broadcast
        const float Bn = x_dbl[(size_t)row * XPROJ_N + DT_RANK + n];
        const float Cn = x_dbl[(size_t)row * XPROJ_N + DT_RANK + D_STATE + n];

        h = __expf(delta * Adn) * h + (delta * u) * Bn;

        float p = h * Cn;                 // butterfly sum over 16 lanes
        p += __shfl_xor(p, 8);
        p += __shfl_xor(p, 4);
        p += __shfl_xor(p, 2);
        p += __shfl_xor(p, 1);

        if (n == 0) {
            const float zv  = xz[(size_t)row * (2 * D_INNER) + D_INNER + d];
            const float sil = zv / (1.0f + __expf(-zv));
            ybuf[(size_t)row * D_INNER + d] = (p + u * Dd) * sil;
        }
    }
}

// ---------------------------------------------------------------------------
// Launch pipeline
// ---------------------------------------------------------------------------
extern "C" void kernel_launch(void* const* d_in, const int* in_sizes, int n_in,
                              void* d_out, int out_size, void* d_ws, size_t ws_size,
                              hipStream_t stream) {
    const float* hs         = (const float*)d_in[0]; // (2,1024,768)
    const float* in_proj_w  = (const float*)d_in[1]; // (3072,768)
    const float* conv_w     = (const float*)d_in[2]; // (1536,4)
    const float* conv_b     = (const float*)d_in[3]; // (1536)
    const float* x_proj_w   = (const float*)d_in[4]; // (80,1536)
    const float* dt_proj_w  = (const float*)d_in[5]; // (1536,48)
    const float* dt_proj_b  = (const float*)d_in[6]; // (1536)
    const float* A_log      = (const float*)d_in[7]; // (1536,16)
    const float* Dv         = (const float*)d_in[8]; // (1536)
    const float* out_proj_w = (const float*)d_in[9]; // (768,1536)
    float* out = (float*)d_out;                      // (2,1024,768)

    float* ws    = (float*)d_ws;
    float* xz    = ws;                               // BL*3072 = 6291456
    float* xconv = xz    + (size_t)BL * 2 * D_INNER; // BL*1536
    float* x_dbl = xconv + (size_t)BL * D_INNER;     // BL*80
    float* draw  = x_dbl + (size_t)BL * XPROJ_N;     // BL*1536
    float* ybuf  = draw  + (size_t)BL * D_INNER;     // BL*1536

    const dim3 blk(128);
    const int mt = BL / 32;   // 64 M-tiles (32 rows per wave)

    // 1) xz = hs @ in_proj_w^T           (2048 x 3072, K=768)
    wmma_gemm_nt<<<dim3((2 * D_INNER + 255) / 256, mt), blk, 0, stream>>>(
        hs, D_MODEL, in_proj_w, D_MODEL, xz, 2 * D_INNER, 2 * D_INNER, D_MODEL);

    // 2) causal depthwise conv + SiLU
    conv_silu_kernel<<<(BL * D_INNER) / 256, 256, 0, stream>>>(
        xz, conv_w, conv_b, xconv);

    // 3) x_dbl = xconv @ x_proj_w^T      (2048 x 80, K=1536)
    wmma_gemm_nt<<<dim3((XPROJ_N + 255) / 256, mt), blk, 0, stream>>>(
        xconv, D_INNER, x_proj_w, D_INNER, x_dbl, XPROJ_N, XPROJ_N, D_INNER);

    // 4) draw = dt @ dt_proj_w^T         (2048 x 1536, K=48; A strided lda=80)
    wmma_gemm_nt<<<dim3((D_INNER + 255) / 256, mt), blk, 0, stream>>>(
        x_dbl, XPROJ_N, dt_proj_w, DT_RANK, draw, D_INNER, D_INNER, DT_RANK);

    // 5) fused selective scan + skip + z-gate
    scan_kernel<<<(B_SZ * D_INNER * 16) / 256, 256, 0, stream>>>(
        xz, xconv, x_dbl, draw, dt_proj_b, A_log, Dv, ybuf);

    // 6) out = ybuf @ out_proj_w^T       (2048 x 768, K=1536)
    wmma_gemm_nt<<<dim3((D_MODEL + 255) / 256, mt), blk, 0, stream>>>(
        ybuf, D_INNER, out_proj_w, D_INNER, out, D_MODEL, D_MODEL, D_INNER);
}